// SumLayer_49916109914250
// MI455X (gfx1250) — compile-verified
//
#include <hip/hip_runtime.h>
#include <hip/hip_bf16.h>
#include <math.h>

// Problem constants (match reference)
#define B_   256
#define P_   128
#define IN_  64
#define OUT_ 64
#define BTILE 64          // b-rows per workgroup
#define PITCH 66          // LDS row pitch in floats (even -> 8B aligned, conflict-free fragment reads)

typedef __attribute__((ext_vector_type(2))) float v2f;
typedef __attribute__((ext_vector_type(8))) float v8f;

// out[b,p,o] = mx[b] + log( sum_i exp(x[b,p,i]-mx[b]) * exp(w[p,o,i]-mw[o]) ) - log( sum_i exp(w[p,o,i]-mw[o]) )
__global__ __launch_bounds__(256) void sumlayer_wmma_kernel(
    const float* __restrict__ x,      // (B, P, IN)
    const float* __restrict__ w,      // (P, OUT, IN)
    float* __restrict__ out)          // (B, P, OUT)
{
    __shared__ float ewT[IN_][PITCH];   // ewT[i][o] = exp(w[p][o][i] - mw[o])  (transposed, B-operand)
    __shared__ float exs[BTILE][PITCH]; // exs[r][i] = exp(x[b0+r][p][i] - mx[r]) (A-operand)
    __shared__ float lsew[OUT_];        // log( sum_i exp(w - mw) ) per o
    __shared__ float mx[BTILE];         // row max of x slab

    const int p    = blockIdx.x;
    const int b0   = blockIdx.y * BTILE;
    const int t    = threadIdx.x;
    const int lane = t & 31;
    const int wave = t >> 5;

    // ---------- Phase 1: weight[p] -> ewT (transposed) + lsew ----------
    {
        const int o = t >> 2;          // 0..63 (one output row per 4-lane group)
        const int q = t & 3;           // quarter of the IN axis
        const float* wrow = w + ((size_t)p * OUT_ + o) * IN_ + q * 16;
        float v[16];
        #pragma unroll
        for (int u = 0; u < 4; ++u) {
            const float4 f = ((const float4*)wrow)[u];
            v[u*4+0] = f.x; v[u*4+1] = f.y; v[u*4+2] = f.z; v[u*4+3] = f.w;
        }
        float m = v[0];
        #pragma unroll
        for (int j = 1; j < 16; ++j) m = fmaxf(m, v[j]);
        m = fmaxf(m, __shfl_xor(m, 1, 32));   // 4-lane group max (groups stay inside a wave)
        m = fmaxf(m, __shfl_xor(m, 2, 32));
        float s = 0.f;
        #pragma unroll
        for (int j = 0; j < 16; ++j) {
            const float e = __expf(v[j] - m);
            s += e;
            ewT[q*16 + j][o] = e;             // transposed store: B[i][o]
        }
        s += __shfl_xor(s, 1, 32);
        s += __shfl_xor(s, 2, 32);
        if (q == 0) lsew[o] = __logf(s);      // mw cancels in the final expression
    }

    // ---------- Phase 2: x slab -> exs + mx ----------
    {
        const int r = t >> 2;          // 0..63 b-row within tile
        const int q = t & 3;
        const float* xrow = x + ((size_t)(b0 + r) * P_ + p) * IN_ + q * 16;
        float v[16];
        #pragma unroll
        for (int u = 0; u < 4; ++u) {
            const float4 f = ((const float4*)xrow)[u];
            v[u*4+0] = f.x; v[u*4+1] = f.y; v[u*4+2] = f.z; v[u*4+3] = f.w;
        }
        float m = v[0];
        #pragma unroll
        for (int j = 1; j < 16; ++j) m = fmaxf(m, v[j]);
        m = fmaxf(m, __shfl_xor(m, 1, 32));
        m = fmaxf(m, __shfl_xor(m, 2, 32));
        #pragma unroll
        for (int j = 0; j < 16; ++j)
            exs[r][q*16 + j] = __expf(v[j] - m);
        if (q == 0) mx[r] = m;
    }

    __syncthreads();

    // ---------- Phase 3: per-wave WMMA GEMM ----------
    // 16 output tiles (4 M x 4 N); wave owns M-tile (wave&3) and N-tiles {2*(wave>>2), +1}
    const int mt    = wave & 3;
    const int nb    = (wave >> 2) * 2;
    const int lrow  = lane & 15;
    const int khalf = (lane >> 4) * 2;    // ISA f32 A/B 16x4 layout: lanes 0-15 hold K={0,1}, 16-31 hold K={2,3}

    v8f acc0 = {};
    v8f acc1 = {};

    #pragma unroll
    for (int k = 0; k < IN_ / 4; ++k) {
        const int koff = k * 4 + khalf;
        v2f a;
        a.x = exs[mt*16 + lrow][koff];
        a.y = exs[mt*16 + lrow][koff + 1];
        v2f bv0;
        bv0.x = ewT[koff    ][nb*16 + lrow];
        bv0.y = ewT[koff + 1][nb*16 + lrow];
        v2f bv1;
        bv1.x = ewT[koff    ][(nb+1)*16 + lrow];
        bv1.y = ewT[koff + 1][(nb+1)*16 + lrow];
        // D = A x B + C ; emits v_wmma_f32_16x16x4_f32
        acc0 = __builtin_amdgcn_wmma_f32_16x16x4_f32(false, a, false, bv0, (short)0, acc0, false, false);
        acc1 = __builtin_amdgcn_wmma_f32_16x16x4_f32(false, a, false, bv1, (short)0, acc1, false, false);
    }

    // ---------- Phase 4: epilogue ----------
    // C/D layout: VGPR r, lanes 0-15 -> M=r, lanes 16-31 -> M=r+8; N = lane&15 within tile
    const int rowadd = (lane >> 4) * 8;
    const int o0 = nb*16 + lrow;
    const int o1 = (nb+1)*16 + lrow;
    const float ls0 = lsew[o0];
    const float ls1 = lsew[o1];
    #pragma unroll
    for (int r = 0; r < 8; ++r) {
        const int row  = mt*16 + r + rowadd;
        const float mr = mx[row];
        const size_t base = ((size_t)(b0 + row) * P_ + p) * OUT_;
        out[base + o0] = mr + __logf(acc0[r]) - ls0;
        out[base + o1] = mr + __logf(acc1[r]) - ls1;
    }
}

extern "C" void kernel_launch(void* const* d_in, const int* in_sizes, int n_in,
                              void* d_out, int out_size, void* d_ws, size_t ws_size,
                              hipStream_t stream) {
    (void)in_sizes; (void)n_in; (void)d_ws; (void)ws_size; (void)out_size;
    const float* x = (const float*)d_in[0];   // (B, P, IN) f32
    const float* w = (const float*)d_in[1];   // (P, OUT, IN) f32
    float* out = (float*)d_out;               // (B, P, OUT) f32

    dim3 grid(P_, B_ / BTILE, 1);             // 128 x 4 workgroups
    dim3 block(256, 1, 1);                    // 8 waves (wave32)
    sumlayer_wmma_kernel<<<grid, block, 0, stream>>>(x, w, out);
}